// Hypergraph_11184094839173
// MI455X (gfx1250) — compile-verified
//
#include <hip/hip_runtime.h>
#include <math.h>

#define N_NODES 50000
#define N_EDGES 600000
#define N_HE    20000
#define HC      128
#define HEADS   4
#define NEG_SLOPE 0.2f
#define EPS_SOFTMAX 1e-16f
#define LN_EPS  1e-5f

typedef __attribute__((ext_vector_type(2))) float v2f;
typedef __attribute__((ext_vector_type(8))) float v8f;
typedef __attribute__((ext_vector_type(4))) unsigned int v4u;
typedef __attribute__((ext_vector_type(8))) int v8i;
typedef __attribute__((ext_vector_type(4))) int v4i;

__device__ __forceinline__ v8f wmma_f32(v2f a, v2f b, v8f c) {
  // V_WMMA_F32_16X16X4_F32: full-fp32 matrix op. Workload is L2/scatter bound,
  // so fp32 WMMA costs nothing vs bf16 and keeps reference numerics exactly.
  return __builtin_amdgcn_wmma_f32_16x16x4_f32(false, a, false, b, (short)0, c,
                                               false, false);
}

// Tensor Data Mover: DMA one 64KB fp32 weight matrix (128x128) into LDS.
// D# built per CDNA5 ISA: 2D tensor, data_size=4B,
// tensor_dim0=tile_dim0=16384, tensor_dim1=tile_dim1=1, type=2, no cluster.
__device__ __forceinline__ void tdm_load_w64k(const float* gptr, unsigned lds_off) {
  unsigned long long ga = (unsigned long long)(uintptr_t)gptr;
  v4u g0;
  g0[0] = 1u;                                        // count=1, user mode
  g0[1] = lds_off;                                   // lds_addr (bytes)
  g0[2] = (unsigned)(ga & 0xFFFFFFFFu);              // global_addr[31:0]
  g0[3] = (unsigned)((ga >> 32) & 0x01FFFFFFu) | (2u << 30);  // addr[56:32] | type=2
  v8i g1;
  g1[0] = (int)(2u << 16);          // workgroup_mask=0, data_size=2 (4B)
  g1[1] = (int)(0x4000u << 16);     // tensor_dim0[15:0]=16384 @ bits 63:48
  g1[2] = (int)(1u << 16);          // tensor_dim0 hi=0, tensor_dim1=1
  g1[3] = (int)(0x4000u << 16);     // tile_dim0=16384 @ bits 127:112
  g1[4] = 1;                        // tile_dim1=1, tile_dim2=0
  g1[5] = 16384;                    // tensor_dim0_stride[31:0]
  g1[6] = (int)(0x4000u << 16);     // dim0_stride hi=0, tensor_dim1_stride=16384
  g1[7] = 0;
  v4i z4 = {0, 0, 0, 0};            // groups 2/3 unused for 2D tile
  v8i z8 = {0, 0, 0, 0, 0, 0, 0, 0};
  __builtin_amdgcn_tensor_load_to_lds(g0, g1, z4, z4, z8, 0);
}

__device__ __forceinline__ float wave_sum32(float v) {
  v += __shfl_xor(v, 1);
  v += __shfl_xor(v, 2);
  v += __shfl_xor(v, 4);
  v += __shfl_xor(v, 8);
  v += __shfl_xor(v, 16);
  return v;
}

__global__ void fill_kernel(float* __restrict__ p, float v, int n) {
  int i = blockIdx.x * blockDim.x + threadIdx.x;
  if (i < n) p[i] = v;
}

// One GEMM k-sweep over a 16-row tile against a 128x128 weight in LDS.
// B operands for all 8 column tiles are loaded into distinct registers first
// (one ds clause + one wait), then 8 independent WMMAs issue back-to-back.
__device__ __forceinline__ void gemm_sweep(const float* __restrict__ arow,
                                           const float* __restrict__ lds_base,
                                           int rsel, int koff, v8f acc[8]) {
  for (int k0 = 0; k0 < HC; k0 += 4) {
    v2f a = *(const v2f*)(arow + k0 + koff);
    v2f bv[8];
#pragma unroll
    for (int ct = 0; ct < 8; ++ct)
      bv[ct] = *(const v2f*)&lds_base[(ct * 16 + rsel) * HC + k0 + koff];
#pragma unroll
    for (int ct = 0; ct < 8; ++ct)
      acc[ct] = wmma_f32(a, bv[ct], acc[ct]);
  }
}

// ---------------------------------------------------------------------------
// Fused projection in two register-bounded passes (64 acc VGPRs live, no spill):
//   pass 1: x@W_K^T + b_K -> alpha_r = sum(x_K * att_r) per head (shuffle reduce)
//   pass 2: x@W_V^T + b_V -> x_V stored
// W_K/W_V staged in LDS once per block via TDM; one wave per 16-node tile.
// ---------------------------------------------------------------------------
__global__ void proj_kernel(const float* __restrict__ x,
                            const float* __restrict__ Wk, const float* __restrict__ bk,
                            const float* __restrict__ Wv, const float* __restrict__ bv,
                            const float* __restrict__ att_r,
                            float* __restrict__ xV, float* __restrict__ alpha_r) {
  extern __shared__ float lds_w[];            // [0..16383]=W_K, [16384..32767]=W_V
  const int lane = threadIdx.x & 31;
  const int tile = blockIdx.x * 8 + (threadIdx.x >> 5);

  if ((threadIdx.x >> 5) == 0) {              // wave 0 issues both TDM loads
    unsigned base = (unsigned)(uintptr_t)&lds_w[0];
    tdm_load_w64k(Wk, base);
    tdm_load_w64k(Wv, base + 65536u);
    __builtin_amdgcn_s_wait_tensorcnt(0);
  }
  __syncthreads();
  if (tile * 16 >= N_NODES) return;

  const int row0  = tile * 16;
  const int rsel  = lane & 15;        // A: M row / B(LDS): N col / D: N col
  const int hi    = lane >> 4;
  const int koff  = hi * 2;           // K offset for A/B operand halves
  const int mbase = hi * 8;           // D: M offset for lane half
  const float* arow = x + (size_t)(row0 + rsel) * HC;

  v8f acc[8];

  // ---- pass 1: K projection -> alpha_r ----
#pragma unroll
  for (int ct = 0; ct < 8; ++ct)
    acc[ct] = (v8f){0.f, 0.f, 0.f, 0.f, 0.f, 0.f, 0.f, 0.f};
  gemm_sweep(arow, lds_w, rsel, koff, acc);
  {
    float acc_alpha[HEADS][8];
#pragma unroll
    for (int h = 0; h < HEADS; ++h)
#pragma unroll
      for (int v = 0; v < 8; ++v) acc_alpha[h][v] = 0.f;
#pragma unroll
    for (int ct = 0; ct < 8; ++ct) {
      const int   col  = ct * 16 + rsel;
      const float bkc  = bk[col];
      const float arc  = att_r[col];
      const int   head = ct >> 1;     // 32 cols per head -> 2 col-tiles per head
#pragma unroll
      for (int v = 0; v < 8; ++v) {
        float p = (acc[ct][v] + bkc) * arc;   // reduce over tile's 16 columns
        p += __shfl_xor(p, 1);
        p += __shfl_xor(p, 2);
        p += __shfl_xor(p, 4);
        p += __shfl_xor(p, 8);
        acc_alpha[head][v] += p;
      }
    }
#pragma unroll
    for (int h = 0; h < HEADS; ++h) {
      if (rsel == h) {
#pragma unroll
        for (int v = 0; v < 8; ++v)
          alpha_r[(size_t)(row0 + mbase + v) * HEADS + h] = acc_alpha[h][v];
      }
    }
  }

  // ---- pass 2: V projection -> x_V ----
#pragma unroll
  for (int ct = 0; ct < 8; ++ct)
    acc[ct] = (v8f){0.f, 0.f, 0.f, 0.f, 0.f, 0.f, 0.f, 0.f};
  gemm_sweep(arow, lds_w + 16384, rsel, koff, acc);
#pragma unroll
  for (int ct = 0; ct < 8; ++ct) {
    const int   col = ct * 16 + rsel;
    const float bvc = bv[col];
#pragma unroll
    for (int v = 0; v < 8; ++v)
      xV[(size_t)(row0 + mbase + v) * HC + col] = acc[ct][v] + bvc;
  }
}

// ---------------------------------------------------------------------------
// Edge pass 1: segment max of leaky_relu(alpha_r[src]) into a_max[dst]
// exact float max via sign-split int/uint atomics (init = -inf)
// ---------------------------------------------------------------------------
__global__ void edge_max_kernel(const int* __restrict__ src, const int* __restrict__ dst,
                                const float* __restrict__ alpha_r, float* __restrict__ amax) {
  int t = blockIdx.x * blockDim.x + threadIdx.x;
  if (t >= N_EDGES * HEADS) return;
  int e = t >> 2, h = t & 3;
  float a = alpha_r[(size_t)src[e] * HEADS + h];
  a = a > 0.f ? a : NEG_SLOPE * a;
  float* p = &amax[(size_t)dst[e] * HEADS + h];
  if (a >= 0.f) atomicMax((int*)p, __float_as_int(a));
  else          atomicMin((unsigned int*)p, __float_as_uint(a));
}

// Edge pass 2: denom[dst] += exp(a - a_max[dst])
__global__ void edge_sum_kernel(const int* __restrict__ src, const int* __restrict__ dst,
                                const float* __restrict__ alpha_r,
                                const float* __restrict__ amax, float* __restrict__ denom) {
  int t = blockIdx.x * blockDim.x + threadIdx.x;
  if (t >= N_EDGES * HEADS) return;
  int e = t >> 2, h = t & 3;
  float a = alpha_r[(size_t)src[e] * HEADS + h];
  a = a > 0.f ? a : NEG_SLOPE * a;
  int idx = dst[e] * HEADS + h;
  atomicAdd(&denom[idx], __expf(a - amax[idx]));
}

// Edge pass 3: agg[dst] += x_V[src] * softmax_weight ; one wave per edge
__global__ void edge_agg_kernel(const int* __restrict__ src, const int* __restrict__ dst,
                                const float* __restrict__ alpha_r,
                                const float* __restrict__ amax, const float* __restrict__ denom,
                                const float* __restrict__ xV, float* __restrict__ agg) {
  const int lane = threadIdx.x & 31;
  const int e = blockIdx.x * 8 + (threadIdx.x >> 5);
  if (e >= N_EDGES) return;
  const int s = src[e], d = dst[e];
  const int head = lane >> 3;                     // 8 lanes (32 floats) per head
  float a = alpha_r[(size_t)s * HEADS + head];
  a = a > 0.f ? a : NEG_SLOPE * a;
  const int idx = d * HEADS + head;
  const float w = __expf(a - amax[idx]) / (denom[idx] + EPS_SOFTMAX);
  const float4 xv = *(const float4*)(xV + (size_t)s * HC + lane * 4);
  float* op = agg + (size_t)d * HC + lane * 4;
  atomicAdd(op + 0, xv.x * w);
  atomicAdd(op + 1, xv.y * w);
  atomicAdd(op + 2, xv.z * w);
  atomicAdd(op + 3, xv.w * w);
}

// LN0: y0 = layernorm(agg + att_r) ; one wave per row
__global__ void ln0_kernel(const float* __restrict__ agg, const float* __restrict__ att_r,
                           const float* __restrict__ g, const float* __restrict__ b,
                           float* __restrict__ y0) {
  const int lane = threadIdx.x & 31;
  const int row = blockIdx.x * 8 + (threadIdx.x >> 5);
  if (row >= N_HE) return;
  float4 v  = *(const float4*)(agg + (size_t)row * HC + lane * 4);
  float4 ar = *(const float4*)(att_r + lane * 4);
  v.x += ar.x; v.y += ar.y; v.z += ar.z; v.w += ar.w;
  float mu = wave_sum32(v.x + v.y + v.z + v.w) * (1.f / HC);
  float dx = v.x - mu, dy = v.y - mu, dz = v.z - mu, dw = v.w - mu;
  float var = wave_sum32(dx * dx + dy * dy + dz * dz + dw * dw) * (1.f / HC);
  float inv = rsqrtf(var + LN_EPS);
  float4 gg = *(const float4*)(g + lane * 4);
  float4 bb = *(const float4*)(b + lane * 4);
  float4 o;
  o.x = dx * inv * gg.x + bb.x;
  o.y = dy * inv * gg.y + bb.y;
  o.z = dz * inv * gg.z + bb.z;
  o.w = dw * inv * gg.w + bb.w;
  *(float4*)(y0 + (size_t)row * HC + lane * 4) = o;
}

// Generic 128-wide WMMA GEMM: out = (relu?)(A @ W^T + bias).
// W staged in LDS via TDM once per block; one wave per 16-row tile.
__global__ void gemm128_kernel(const float* __restrict__ A, const float* __restrict__ W,
                               const float* __restrict__ bias, float* __restrict__ out,
                               int M, int do_relu) {
  extern __shared__ float lds_w[];            // 16384 floats = W (64KB)
  const int lane = threadIdx.x & 31;
  const int tile = blockIdx.x * 8 + (threadIdx.x >> 5);

  if ((threadIdx.x >> 5) == 0) {
    tdm_load_w64k(W, (unsigned)(uintptr_t)&lds_w[0]);
    __builtin_amdgcn_s_wait_tensorcnt(0);
  }
  __syncthreads();
  if (tile * 16 >= M) return;

  const int row0  = tile * 16;
  const int rsel  = lane & 15;
  const int koff  = (lane >> 4) * 2;
  const int mbase = (lane >> 4) * 8;

  v8f acc[8];
#pragma unroll
  for (int ct = 0; ct < 8; ++ct)
    acc[ct] = (v8f){0.f, 0.f, 0.f, 0.f, 0.f, 0.f, 0.f, 0.f};

  const float* arow = A + (size_t)(row0 + rsel) * HC;
  gemm_sweep(arow, lds_w, rsel, koff, acc);

#pragma unroll
  for (int ct = 0; ct < 8; ++ct) {
    const int col = ct * 16 + rsel;
    const float bc = bias[col];
#pragma unroll
    for (int v = 0; v < 8; ++v) {
      float val = acc[ct][v] + bc;
      if (do_relu) val = fmaxf(val, 0.f);
      out[(size_t)(row0 + mbase + v) * HC + col] = val;
    }
  }
}

// LN1: out = layernorm(y0 + relu(ff))
__global__ void ln1_kernel(const float* __restrict__ y0, const float* __restrict__ ff,
                           const float* __restrict__ g, const float* __restrict__ b,
                           float* __restrict__ out) {
  const int lane = threadIdx.x & 31;
  const int row = blockIdx.x * 8 + (threadIdx.x >> 5);
  if (row >= N_HE) return;
  float4 v = *(const float4*)(y0 + (size_t)row * HC + lane * 4);
  float4 f = *(const float4*)(ff + (size_t)row * HC + lane * 4);
  v.x += fmaxf(f.x, 0.f); v.y += fmaxf(f.y, 0.f);
  v.z += fmaxf(f.z, 0.f); v.w += fmaxf(f.w, 0.f);
  float mu = wave_sum32(v.x + v.y + v.z + v.w) * (1.f / HC);
  float dx = v.x - mu, dy = v.y - mu, dz = v.z - mu, dw = v.w - mu;
  float var = wave_sum32(dx * dx + dy * dy + dz * dz + dw * dw) * (1.f / HC);
  float inv = rsqrtf(var + LN_EPS);
  float4 gg = *(const float4*)(g + lane * 4);
  float4 bb = *(const float4*)(b + lane * 4);
  float4 o;
  o.x = dx * inv * gg.x + bb.x;
  o.y = dy * inv * gg.y + bb.y;
  o.z = dz * inv * gg.z + bb.z;
  o.w = dw * inv * gg.w + bb.w;
  *(float4*)(out + (size_t)row * HC + lane * 4) = o;
}

extern "C" void kernel_launch(void* const* d_in, const int* in_sizes, int n_in,
                              void* d_out, int out_size, void* d_ws, size_t ws_size,
                              hipStream_t stream) {
  (void)in_sizes; (void)n_in; (void)out_size; (void)ws_size;
  const float* x     = (const float*)d_in[0];
  const int*   src   = (const int*)d_in[1];
  const int*   dst   = (const int*)d_in[2];
  const float* W_K   = (const float*)d_in[3];
  const float* b_K   = (const float*)d_in[4];
  const float* W_V   = (const float*)d_in[5];
  const float* b_V   = (const float*)d_in[6];
  const float* att_r = (const float*)d_in[7];
  const float* W1    = (const float*)d_in[8];
  const float* b1    = (const float*)d_in[9];
  const float* W2    = (const float*)d_in[10];
  const float* b2    = (const float*)d_in[11];
  const float* ln0_g = (const float*)d_in[12];
  const float* ln0_b = (const float*)d_in[13];
  const float* ln1_g = (const float*)d_in[14];
  const float* ln1_b = (const float*)d_in[15];
  float* out = (float*)d_out;

  float* ws = (float*)d_ws;
  float* alpha_r = ws;                       // 200,000
  float* xV      = ws + 200000;              // 6,400,000
  float* amax    = ws + 6600000;             // 80,000
  float* denom   = ws + 6680000;             // 80,000
  float* agg     = ws + 6760000;             // 2,560,000
  float* y0      = ws + 9320000;             // 2,560,000
  float* hbuf    = ws + 11880000;            // 2,560,000
  float* ffbuf   = ws + 14440000;            // 2,560,000  (total 68 MB)

  const float NEG_INF = -__builtin_inff();
  fill_kernel<<<(80000 + 255) / 256, 256, 0, stream>>>(amax, NEG_INF, 80000);
  fill_kernel<<<(80000 + 255) / 256, 256, 0, stream>>>(denom, 0.f, 80000);
  fill_kernel<<<(2560000 + 255) / 256, 256, 0, stream>>>(agg, 0.f, 2560000);

  // projections + alpha_r (WMMA + TDM-staged weights): 3125 tiles, 8 waves/block
  proj_kernel<<<(3125 + 7) / 8, 256, 131072, stream>>>(x, W_K, b_K, W_V, b_V,
                                                       att_r, xV, alpha_r);

  const int eh_blocks = (N_EDGES * HEADS + 255) / 256;
  edge_max_kernel<<<eh_blocks, 256, 0, stream>>>(src, dst, alpha_r, amax);
  edge_sum_kernel<<<eh_blocks, 256, 0, stream>>>(src, dst, alpha_r, amax, denom);
  edge_agg_kernel<<<N_EDGES / 8, 256, 0, stream>>>(src, dst, alpha_r, amax, denom,
                                                   xV, agg);

  ln0_kernel<<<N_HE / 8, 256, 0, stream>>>(agg, att_r, ln0_g, ln0_b, y0);

  // MLP (WMMA + TDM-staged weights): 1250 tiles, 8 waves/block
  gemm128_kernel<<<(1250 + 7) / 8, 256, 65536, stream>>>(y0, W1, b1, hbuf, N_HE, 1);
  gemm128_kernel<<<(1250 + 7) / 8, 256, 65536, stream>>>(hbuf, W2, b2, ffbuf, N_HE, 0);

  ln1_kernel<<<N_HE / 8, 256, 0, stream>>>(y0, ffbuf, ln1_g, ln1_b, out);
}